// development_layer_49795850830418
// MI455X (gfx1250) — compile-verified
//
#include <hip/hip_runtime.h>

typedef __attribute__((ext_vector_type(2))) float v2f;
typedef __attribute__((ext_vector_type(8))) float v8f;

#define NN   64
#define TT   1024
#define DD   8
#define CC   6
#define CH   8            // time chunks (parallelism over T)
#define LCH  128          // ceil((TT-1)/CH)
#define ROW  18           // padded LDS row stride in floats (bank-conflict free, b64-alignable)
#define MATW (16*ROW)
#define TORD 12           // Taylor order for expm of the scaled matrix (norm <= 1)
#define MAXSQ 14          // matches reference MAX_SQUARINGS

// lane ^ m shuffle via DS_SWIZZLE_B32, group-of-32 mode: and=0x1f, or=0, xor=m
#define SWZ_XOR(v, m) __int_as_float(__builtin_amdgcn_ds_swizzle(__float_as_int(v), (((m) << 10) | 0x1f)))

// ---- WMMA fragment helpers -------------------------------------------------
// D/C layout (f32 16x16): reg r, lane l -> element (m = r + 8*(l>>4), n = l&15)
// A layout  (f32 16x4)  : lane l, comp v -> A(m = l&15, k = 2*(l>>4) + v)
// B layout  (f32 4x16)  : lane l, comp v -> B(k = 2*(l>>4) + v, n = l&15)

__device__ __forceinline__ void load_afr(const float* M, int lo, int hi, v2f a[4]) {
#pragma unroll
  for (int kk = 0; kk < 4; ++kk) {
    const int k = 4 * kk + 2 * hi;
    a[kk].x = M[lo * ROW + k];
    a[kk].y = M[lo * ROW + k + 1];
  }
}

__device__ __forceinline__ void load_bfr(const float* M, int lo, int hi, v2f b[4]) {
#pragma unroll
  for (int kk = 0; kk < 4; ++kk) {
    const int k = 4 * kk + 2 * hi;
    b[kk].x = M[k * ROW + lo];
    b[kk].y = M[(k + 1) * ROW + lo];
  }
}

__device__ __forceinline__ v8f wmma4(const v2f a[4], const v2f b[4]) {
  v8f acc = 0.0f;
#pragma unroll
  for (int kk = 0; kk < 4; ++kk) {
    // full 16x16x16 f32 matmul as 4 chained K=4 WMMAs
    acc = __builtin_amdgcn_wmma_f32_16x16x4_f32(false, a[kk], false, b[kk],
                                                (short)0, acc, false, false);
  }
  return acc;
}

// ---- Kernel 1: per (n, c, chunk) ordered product of expm(G_t) --------------
__global__ __launch_bounds__(32) void dl_expm_chain_k1(
    const float* __restrict__ x,    // (N, T, D)
    const float* __restrict__ Are,  // (D, C, 8, 8)
    const float* __restrict__ Aim,  // (D, C, 8, 8)
    float* __restrict__ part)       // (N*C*CH, 256) partial 16x16 products
{
  __shared__ __align__(16) float BG[MATW];
  __shared__ __align__(16) float BY[MATW];
  __shared__ __align__(16) float BP[MATW];

  const int wid = blockIdx.x;
  const int ch  = wid % CH;
  const int c   = (wid / CH) % CC;
  const int n   = wid / (CH * CC);
  const int lane = threadIdx.x;
  const int lo = lane & 15, hi = lane >> 4;
  const int jj = lo & 7;
  const int right = (lo >> 3) & 1;

  // Skew-Hermitian projection + complex->real 16x16 embedding coefficients.
  // A = 0.5*(A - A^H): Re part antisym(Are), Im part sym(Aim).
  // Embedding quadrants: TL/BR = Re, BL = +Im, TR = -Im.
  float V[DD][8];
#pragma unroll
  for (int d = 0; d < DD; ++d) {
    const float* are = Are + ((d * CC + c) << 6);
    const float* aim = Aim + ((d * CC + c) << 6);
#pragma unroll
    for (int r = 0; r < 8; ++r) {
      const float reV = 0.5f * (are[r * 8 + jj] - are[jj * 8 + r]);
      const float imV = 0.5f * (aim[r * 8 + jj] + aim[jj * 8 + r]);
      V[d][r] = (hi == right) ? reV : (hi ? imV : -imV);
    }
  }

  // P = I
#pragma unroll
  for (int r = 0; r < 8; ++r) {
    const int m = r + 8 * hi;
    BP[m * ROW + lo] = (m == lo) ? 1.0f : 0.0f;
  }

  const int t0 = ch * LCH;
  const int t1 = min(t0 + LCH, TT - 1);

  float xc[DD];
  {
    const float* xr = x + (size_t)(n * TT + t0) * DD;
#pragma unroll
    for (int d = 0; d < DD; ++d) xc[d] = xr[d];
  }
  __syncthreads();

  for (int t = t0; t < t1; ++t) {
    // path increment dX[t] and G = sum_d dX_d * S_d (this lane's 8 elements)
    float dxv[DD];
    const float* xr = x + (size_t)(n * TT + t + 1) * DD;
#pragma unroll
    for (int d = 0; d < DD; ++d) { const float xv = xr[d]; dxv[d] = xv - xc[d]; xc[d] = xv; }

    float g[8];
#pragma unroll
    for (int r = 0; r < 8; ++r) {
      float s = 0.0f;
#pragma unroll
      for (int d = 0; d < DD; ++d) s = fmaf(dxv[d], V[d][r], s);
      g[r] = s;
    }

    // induced 1-norm of the 16x16 embedding: max_n sum_m |G[m][n]|
    float cs = 0.0f;
#pragma unroll
    for (int r = 0; r < 8; ++r) cs += fabsf(g[r]);
    cs += SWZ_XOR(cs, 16);                 // combine row halves of this column
    float nm = cs;
    nm = fmaxf(nm, SWZ_XOR(nm, 1));
    nm = fmaxf(nm, SWZ_XOR(nm, 2));
    nm = fmaxf(nm, SWZ_XOR(nm, 4));
    nm = fmaxf(nm, SWZ_XOR(nm, 8));        // wave-uniform max over columns

    int s_cnt = 0;
    if (nm > 1.0f) s_cnt = (int)ceilf(log2f(nm));
    s_cnt = min(max(s_cnt, 0), MAXSQ);
    const float sc = exp2f(-(float)s_cnt);

    // store scaled G, seed Horner iterate Y = I + G/TORD
#pragma unroll
    for (int r = 0; r < 8; ++r) {
      const int m = r + 8 * hi;
      const float gs = g[r] * sc;
      BG[m * ROW + lo] = gs;
      BY[m * ROW + lo] = gs * (1.0f / (float)TORD) + ((m == lo) ? 1.0f : 0.0f);
    }
    __syncthreads();

    v2f aG[4];
    load_afr(BG, lo, hi, aG);

    // Taylor-12 Horner: Y = I + (G @ Y)/k, k = 11..1
#pragma unroll
    for (int k = TORD - 1; k >= 1; --k) {
      v2f bY[4];
      load_bfr(BY, lo, hi, bY);
      const v8f acc = wmma4(aG, bY);
      __syncthreads();
#pragma unroll
      for (int r = 0; r < 8; ++r) {
        const int m = r + 8 * hi;
        BY[m * ROW + lo] = acc[r] * (1.0f / (float)k) + ((m == lo) ? 1.0f : 0.0f);
      }
      __syncthreads();
    }

    // repeated squaring (wave-uniform count)
    for (int q = 0; q < s_cnt; ++q) {
      v2f aY[4], bY[4];
      load_afr(BY, lo, hi, aY);
      load_bfr(BY, lo, hi, bY);
      const v8f acc = wmma4(aY, bY);
      __syncthreads();
#pragma unroll
      for (int r = 0; r < 8; ++r) BY[(r + 8 * hi) * ROW + lo] = acc[r];
      __syncthreads();
    }

    // ordered product: P = P @ E_t
    {
      v2f aP[4], bY[4];
      load_afr(BP, lo, hi, aP);
      load_bfr(BY, lo, hi, bY);
      const v8f acc = wmma4(aP, bY);
      __syncthreads();
#pragma unroll
      for (int r = 0; r < 8; ++r) BP[(r + 8 * hi) * ROW + lo] = acc[r];
      __syncthreads();
    }
  }

  float* dst = part + (size_t)wid * 256;
#pragma unroll
  for (int r = 0; r < 8; ++r) {
    const int m = r + 8 * hi;
    dst[m * 16 + lo] = BP[m * ROW + lo];
  }
}

// ---- Kernel 2: reduce CH chunk partials in order, emit complex64 -----------
__global__ __launch_bounds__(32) void dl_expm_chain_k2(
    const float* __restrict__ part, float* __restrict__ out)
{
  __shared__ __align__(16) float BP[MATW];
  __shared__ __align__(16) float BE[MATW];

  const int c = blockIdx.x % CC;
  const int n = blockIdx.x / CC;
  const int lane = threadIdx.x;
  const int lo = lane & 15, hi = lane >> 4;

#pragma unroll
  for (int r = 0; r < 8; ++r) {
    const int m = r + 8 * hi;
    BP[m * ROW + lo] = (m == lo) ? 1.0f : 0.0f;
  }
  __syncthreads();

  for (int ch = 0; ch < CH; ++ch) {
    const float* src = part + (size_t)((n * CC + c) * CH + ch) * 256;
#pragma unroll
    for (int j = 0; j < 8; ++j) {
      const int idx = j * 32 + lane;
      BE[(idx >> 4) * ROW + (idx & 15)] = src[idx];
    }
    __syncthreads();
    v2f aP[4], bE[4];
    load_afr(BP, lo, hi, aP);
    load_bfr(BE, lo, hi, bE);
    const v8f acc = wmma4(aP, bE);
    __syncthreads();
#pragma unroll
    for (int r = 0; r < 8; ++r) BP[(r + 8 * hi) * ROW + lo] = acc[r];
    __syncthreads();
  }

  // embedding -> complex64: re = P[i][j], im = P[i+8][j]
#pragma unroll
  for (int e = 0; e < 2; ++e) {
    const int idx = lane * 2 + e;      // 0..63
    const int i = idx >> 3, j = idx & 7;
    float* o = out + 2 * ((size_t)((n * CC + c) << 6) + idx);
    o[0] = BP[i * ROW + j];
    o[1] = BP[(i + 8) * ROW + j];
  }
}

extern "C" void kernel_launch(void* const* d_in, const int* in_sizes, int n_in,
                              void* d_out, int out_size, void* d_ws, size_t ws_size,
                              hipStream_t stream) {
  (void)in_sizes; (void)n_in; (void)out_size; (void)ws_size;
  const float* x   = (const float*)d_in[0];
  const float* Are = (const float*)d_in[1];
  const float* Aim = (const float*)d_in[2];
  float* part = (float*)d_ws;  // N*C*CH * 256 floats = 3 MB scratch

  dl_expm_chain_k1<<<NN * CC * CH, 32, 0, stream>>>(x, Are, Aim, part);
  dl_expm_chain_k2<<<NN * CC, 32, 0, stream>>>(part, (float*)d_out);
}